// EquiPINE_36584531427734
// MI455X (gfx1250) — compile-verified
//
#include <hip/hip_runtime.h>

typedef float v2f __attribute__((ext_vector_type(2)));
typedef float v8f __attribute__((ext_vector_type(8)));

#define B_ 64
#define N_ 1024
#define D_ 16
#define L_ 32
#define M_ 128      // NP*PS
#define H_ 256
#define O_ 128
#define K2M_ 256    // 2*M
#define NCHUNK 128
#define NCHUNKS (N_ / NCHUNK)   // 8

__device__ __forceinline__ float fast_sigmoid(float t) {
    // 1 / (1 + exp(-t)) : v_mul + v_exp_f32 + v_add + v_rcp_f32
    return __builtin_amdgcn_rcpf(1.0f + __expf(-t));
}

// ---------------------------------------------------------------------------
// Kernel 1: w[m] = U[np,ps] * dot(A[np,:], P[:])  (rank-1 collapse), both branches.
// w01 layout: [0..127] = branch0, [128..255] = branch1.
// ---------------------------------------------------------------------------
__global__ void prep_w(const float* __restrict__ P0, const float* __restrict__ U0,
                       const float* __restrict__ A0, const float* __restrict__ P1,
                       const float* __restrict__ U1, const float* __restrict__ A1,
                       float* __restrict__ w01) {
    int t  = threadIdx.x;          // 0..255
    int br = t >> 7;
    int m  = t & (M_ - 1);
    const float* P = br ? P1 : P0;
    const float* U = br ? U1 : U0;
    const float* A = br ? A1 : A0;
    int np = m >> 3;               // m / PS
    float dot = 0.0f;
#pragma unroll
    for (int l = 0; l < L_; ++l) dot += A[np * L_ + l] * P[l];
    w01[t] = U[m] * dot;
}

// ---------------------------------------------------------------------------
// Kernel 2: per (b, n-chunk) partial of z.
// Phase A: each thread reduces one x-row (D=16) to (max,min) -> LDS.
// Phase B: thread (br,m) accumulates sum_n sigmoid(max(w*mx, w*mn) + V).
//   (monotonicity of sigmoid makes this exactly equal to max_d sigmoid(...))
// ---------------------------------------------------------------------------
__device__ __forceinline__ void mm4(float4 r, float& mx, float& mn) {
    mx = fmaxf(mx, fmaxf(fmaxf(r.x, r.y), fmaxf(r.z, r.w)));
    mn = fminf(mn, fminf(fminf(r.x, r.y), fminf(r.z, r.w)));
}

__global__ void branch_partial(const float* __restrict__ x0,
                               const float* __restrict__ x1,
                               const float* __restrict__ V0,
                               const float* __restrict__ V1,
                               const float* __restrict__ w01,
                               float* __restrict__ partial /* [B][NCHUNKS][256] */) {
    __shared__ float smax[2][NCHUNK];
    __shared__ float smin[2][NCHUNK];

    int b     = blockIdx.x;        // 0..63
    int chunk = blockIdx.y;        // 0..7
    int t     = threadIdx.x;       // 0..255
    int br    = t >> 7;
    int lid   = t & (NCHUNK - 1);  // row-in-chunk (phase A) / m (phase B)

    // ---- Phase A: row max/min over D=16 (64B contiguous -> 4x float4) ----
    const float* x = br ? x1 : x0;
    const float4* row =
        (const float4*)(x + ((size_t)b * N_ + (size_t)chunk * NCHUNK + lid) * D_);
    float4 r0 = row[0], r1 = row[1], r2 = row[2], r3 = row[3];
    float mx = r0.x, mn = r0.x;
    mm4(r0, mx, mn); mm4(r1, mx, mn); mm4(r2, mx, mn); mm4(r3, mx, mn);
    smax[br][lid] = mx;
    smin[br][lid] = mn;
    __syncthreads();

    // ---- Phase B: accumulate over the 128 n's in this chunk ----
    float w = w01[t];
    float V = (br ? V1 : V0)[lid];
    float acc = 0.0f;
#pragma unroll 4
    for (int n = 0; n < NCHUNK; ++n) {
        float a  = w * smax[br][n];
        float bb = w * smin[br][n];
        acc += fast_sigmoid(fmaxf(a, bb) + V);
    }
    partial[((size_t)b * NCHUNKS + chunk) * 256 + t] = acc;
}

// ---------------------------------------------------------------------------
// Kernel 3: z[b][c] = sum over chunks (fixed order -> deterministic).
// c layout already matches concat([branch0, branch1]).
// ---------------------------------------------------------------------------
__global__ void reduce_z(const float* __restrict__ partial, float* __restrict__ z) {
    int i = blockIdx.x * blockDim.x + threadIdx.x;   // 0 .. 64*256-1
    int b = i >> 8;
    int c = i & 255;
    float s = 0.0f;
#pragma unroll
    for (int k = 0; k < NCHUNKS; ++k)
        s += partial[((size_t)b * NCHUNKS + k) * 256 + c];
    z[i] = s;
}

// ---------------------------------------------------------------------------
// Kernel 4: h = sigmoid(z @ W_h^T), via V_WMMA_F32_16X16X4_F32.
// One wave32 per 16x16 tile. A = z rows, B[k][j] = W_h[j][k] (both read as
// contiguous float2 at k + 2*(lane>>4), per the ISA 16x4 / 4x16 f32 layouts).
// ---------------------------------------------------------------------------
__global__ void mlp_hidden(const float* __restrict__ z, const float* __restrict__ Wh,
                           float* __restrict__ h) {
    int lane = threadIdx.x;            // 0..31
    int tile = blockIdx.x;             // 4 row-tiles x 16 col-tiles = 64
    int tb   = (tile >> 4) << 4;       // b base
    int tj   = (tile & 15) << 4;       // hidden-col base
    int half = lane >> 4;
    int lid  = lane & 15;

    const float* arow = z  + (size_t)(tb + lid) * K2M_;
    const float* brow = Wh + (size_t)(tj + lid) * K2M_;

    v8f c = {};
#pragma unroll 8
    for (int k = 0; k < K2M_; k += 4) {
        v2f a  = *(const v2f*)(arow + k + 2 * half);
        v2f bb = *(const v2f*)(brow + k + 2 * half);
        c = __builtin_amdgcn_wmma_f32_16x16x4_f32(false, a, false, bb,
                                                  (short)0, c, false, false);
    }
#pragma unroll
    for (int v = 0; v < 8; ++v) {
        int row = tb + v + half * 8;   // C/D layout: lanes 16-31 hold M = v+8
        int col = tj + lid;
        h[(size_t)row * H_ + col] = fast_sigmoid(c[v]);
    }
}

// ---------------------------------------------------------------------------
// Kernel 5: out = h @ C_w^T + C_b, same WMMA scheme. 4 x 8 tiles = 32 waves.
// ---------------------------------------------------------------------------
__global__ void mlp_out(const float* __restrict__ h, const float* __restrict__ Cw,
                        const float* __restrict__ Cb, float* __restrict__ out) {
    int lane = threadIdx.x;
    int tile = blockIdx.x;             // 4 row-tiles x 8 col-tiles = 32
    int tb   = (tile >> 3) << 4;
    int to   = (tile & 7) << 4;
    int half = lane >> 4;
    int lid  = lane & 15;

    const float* arow = h  + (size_t)(tb + lid) * H_;
    const float* brow = Cw + (size_t)(to + lid) * H_;

    v8f c = {};
#pragma unroll 8
    for (int k = 0; k < H_; k += 4) {
        v2f a  = *(const v2f*)(arow + k + 2 * half);
        v2f bb = *(const v2f*)(brow + k + 2 * half);
        c = __builtin_amdgcn_wmma_f32_16x16x4_f32(false, a, false, bb,
                                                  (short)0, c, false, false);
    }
#pragma unroll
    for (int v = 0; v < 8; ++v) {
        int row = tb + v + half * 8;
        int col = to + lid;
        out[(size_t)row * O_ + col] = c[v] + Cb[col];
    }
}

// ---------------------------------------------------------------------------
extern "C" void kernel_launch(void* const* d_in, const int* in_sizes, int n_in,
                              void* d_out, int out_size, void* d_ws, size_t ws_size,
                              hipStream_t stream) {
    const float* x0 = (const float*)d_in[0];
    const float* x1 = (const float*)d_in[1];
    const float* P0 = (const float*)d_in[2];
    const float* U0 = (const float*)d_in[3];
    const float* A0 = (const float*)d_in[4];
    const float* V0 = (const float*)d_in[5];
    const float* P1 = (const float*)d_in[6];
    const float* U1 = (const float*)d_in[7];
    const float* A1 = (const float*)d_in[8];
    const float* V1 = (const float*)d_in[9];
    const float* Wh = (const float*)d_in[10];
    const float* Cw = (const float*)d_in[11];
    const float* Cb = (const float*)d_in[12];
    float* out = (float*)d_out;

    float* ws      = (float*)d_ws;
    float* w01     = ws;                                 // 256
    float* partial = w01 + 256;                          // 64*8*256 = 131072
    float* z       = partial + (size_t)B_ * NCHUNKS * 256; // 16384
    float* h       = z + (size_t)B_ * K2M_;              // 16384
    // total ws use: ~656 KB of f32

    prep_w<<<1, 256, 0, stream>>>(P0, U0, A0, P1, U1, A1, w01);

    dim3 grid(B_, NCHUNKS);
    branch_partial<<<grid, 256, 0, stream>>>(x0, x1, V0, V1, w01, partial);

    reduce_z<<<(B_ * 256) / 256, 256, 0, stream>>>(partial, z);

    mlp_hidden<<<64, 32, 0, stream>>>(z, Wh, h);
    mlp_out<<<32, 32, 0, stream>>>(h, Cw, Cb, out);
}